// embedding_ball_pn_64037962383754
// MI455X (gfx1250) — compile-verified
//
#include <hip/hip_runtime.h>
#include <math.h>

typedef __attribute__((ext_vector_type(2))) float v2f;
typedef __attribute__((ext_vector_type(8))) float v8f;

#define NPTS   256
#define NCLOUD 128
#define KNN    16
#define TOTAL_PTS (NCLOUD * NPTS)   // 32768 points per stage
#define GEMM_BLOCKS (TOTAL_PTS / 8) // 8 points (waves) per block

__device__ __forceinline__ float gelu_erf(float x) {
    return 0.5f * x * (1.0f + erff(x * 0.70710678118654752440f));
}

// ---------------------------------------------------------------------------
// K0: (4,6,32,256) -> h0[cloud=4*32][s=256][c=6]
// ---------------------------------------------------------------------------
__global__ void prep_kernel(const float* __restrict__ x, float* __restrict__ h0) {
    int tid = blockIdx.x * blockDim.x + threadIdx.x;
    if (tid >= 4 * 32 * 256 * 6) return;
    int c = tid % 6;
    int s = (tid / 6) % 256;
    int cloud = tid / (6 * 256);
    int b = cloud >> 5, p = cloud & 31;
    h0[tid] = x[((b * 6 + c) * 32 + p) * 256 + s];
}

// ---------------------------------------------------------------------------
// K1: kNN per cloud. One block per cloud; cloud cached in LDS.
// Selection replicates jax.lax.top_k stable ordering (strict >, ascending j).
// ---------------------------------------------------------------------------
template <int C_IN>
__global__ __launch_bounds__(256) void knn_kernel(const float* __restrict__ h,
                                                  int* __restrict__ idx) {
    constexpr int CP = C_IN + 1;               // pad -> odd stride, no bank conflicts
    __shared__ float xs[NPTS * CP];
    __shared__ float sq[NPTS];
    __shared__ float dist[16 * NPTS];          // 16-row chunks, 16 KB
    const int cloud = blockIdx.x;
    const int tid = threadIdx.x;
    const float* hc = h + (size_t)cloud * NPTS * C_IN;

    for (int e = tid; e < NPTS * C_IN; e += 256) {
        int n = e / C_IN, c = e % C_IN;
        xs[n * CP + c] = hc[e];
    }
    __syncthreads();
    {
        float s = 0.f;
        for (int c = 0; c < C_IN; ++c) { float v = xs[tid * CP + c]; s += v * v; }
        sq[tid] = s;
    }
    __syncthreads();

    for (int r0 = 0; r0 < NPTS; r0 += 16) {
        for (int rr = 0; rr < 16; ++rr) {
            int i = r0 + rr;                   // uniform -> broadcast reads
            int j = tid;
            float d = 0.f;
            for (int c = 0; c < C_IN; ++c) d += xs[i * CP + c] * xs[j * CP + c];
            dist[rr * NPTS + j] = 2.f * d - sq[i] - sq[j];
        }
        __syncthreads();
        if (tid < 16) {
            unsigned chosen[8] = {0, 0, 0, 0, 0, 0, 0, 0};
            int row = r0 + tid;
            int* out = idx + ((size_t)cloud * NPTS + row) * KNN;
            for (int kk = 0; kk < KNN; ++kk) {
                float best = -3.4e38f; int bestj = 0;
                for (int j = 0; j < NPTS; ++j) {
                    if ((chosen[j >> 5] >> (j & 31)) & 1u) continue;
                    float v = dist[tid * NPTS + j];
                    if (v > best) { best = v; bestj = j; }
                }
                chosen[bestj >> 5] |= 1u << (bestj & 31);
                out[kk] = bestj;
            }
        }
        __syncthreads();
    }
}

// ---------------------------------------------------------------------------
// K2/K3: fused EdgeConv GEMM via V_WMMA_F32_16X16X4_F32.
// One wave == one point (16 neighbors == 16 M-rows). Register-blocked over
// NT=4 o-tiles (16x64 output per wave): A loaded once per K-step, 4 WMMAs.
// PASSA=true : accumulate per-channel sum/sumsq -> deterministic partials.
// PASSA=false: BN affine + erf-GELU + max over k -> hout[point][O].
// ---------------------------------------------------------------------------
template <int C_IN, int O, bool PASSA>
__global__ __launch_bounds__(256) void edge_gemm_kernel(
        const float* __restrict__ h, const int* __restrict__ idx,
        const float* __restrict__ W, const float* __restrict__ bias,
        const float* __restrict__ scale, const float* __restrict__ shift,
        float* __restrict__ partials, float* __restrict__ hout) {
    constexpr int C2  = 2 * C_IN;
    constexpr int C2P = C2 + 4;                // pad: stride%64==4 dwords -> conflict-free
    constexpr int NT  = (O >= 64) ? 4 : (O / 16);  // o-tiles per register block
    constexpr int OG  = 16 * NT;                   // cols per register block
    __shared__ float feat[8 * 16 * C2P];       // <=149504B total with bstats (stage 5)
    __shared__ float bstats[8 * 2 * O];        // per-wave deterministic slots

    const int tid  = threadIdx.x;
    const int wv   = tid >> 5, lane = tid & 31;
    const int half = lane >> 4, lrow = lane & 15;
    const int pbase = blockIdx.x * 8;

    // Cooperative feat staging: feat[w][kk][c] = concat(nbr - ctr, ctr)
    for (int e = tid; e < 8 * 16 * C2; e += 256) {
        int c  = e % C2;
        int kk = (e / C2) & 15;
        int w2 = e / (C2 * 16);
        int p  = pbase + w2;
        int cloud = p >> 8;
        float v;
        if (c < C_IN) {
            int j = idx[(size_t)p * KNN + kk];
            v = h[((size_t)cloud * NPTS + j) * C_IN + c] - h[(size_t)p * C_IN + c];
        } else {
            v = h[(size_t)p * C_IN + (c - C_IN)];
        }
        feat[(w2 * 16 + kk) * C2P + c] = v;
    }
    __syncthreads();

    const int p = pbase + wv;
    const float* arow = &feat[(wv * 16 + lrow) * C2P + 2 * half];
    const v8f vzero = {0.f, 0.f, 0.f, 0.f, 0.f, 0.f, 0.f, 0.f};

    for (int og = 0; og < O; og += OG) {
        v8f acc[NT];
        #pragma unroll
        for (int t = 0; t < NT; ++t) acc[t] = vzero;

        const float* wbase = W + (size_t)(og + lrow) * C2 + 2 * half;
        #pragma unroll 2
        for (int k0 = 0; k0 < C2; k0 += 4) {
            v2f a = *(const v2f*)(arow + k0);    // A: feat[kk=lrow][k0+2*half+{0,1}]
            #pragma unroll
            for (int t = 0; t < NT; ++t) {       // B: W[og+16t+lrow][k0+2*half+{0,1}]
                v2f bb = *(const v2f*)(wbase + (size_t)(16 * t) * C2 + k0);
                acc[t] = __builtin_amdgcn_wmma_f32_16x16x4_f32(
                    false, a, false, bb, (short)0, acc[t], false, false);
            }
        }

        #pragma unroll
        for (int t = 0; t < NT; ++t) {
            const int ot = og + 16 * t;
            float bv = bias[ot + lrow];
            if (PASSA) {
                float s1 = 0.f, s2 = 0.f;
                #pragma unroll
                for (int r = 0; r < 8; ++r) { float y = acc[t][r] + bv; s1 += y; s2 += y * y; }
                s1 += __shfl_xor(s1, 16, 32);    // combine kk halves (rows r vs r+8)
                s2 += __shfl_xor(s2, 16, 32);
                if (half == 0) {
                    bstats[wv * 2 * O + ot + lrow]     = s1;
                    bstats[wv * 2 * O + O + ot + lrow] = s2;
                }
            } else {
                float sc = scale[ot + lrow], sh = shift[ot + lrow];
                float m = -3.4e38f;
                #pragma unroll
                for (int r = 0; r < 8; ++r) {
                    float y = (acc[t][r] + bv) * sc + sh;   // BN affine (mean/var folded)
                    m = fmaxf(m, gelu_erf(y));
                }
                m = fmaxf(m, __shfl_xor(m, 16, 32));        // max over all 16 kk
                if (half == 0) hout[(size_t)p * O + ot + lrow] = m;
            }
        }
    }
    if (PASSA) {
        __syncthreads();
        for (int e = tid; e < 2 * O; e += 256) {
            float a2 = 0.f;
            #pragma unroll
            for (int w = 0; w < 8; ++w) a2 += bstats[w * 2 * O + e];  // fixed order
            partials[(size_t)blockIdx.x * (2 * O) + e] = a2;
        }
    }
}

// ---------------------------------------------------------------------------
// K2b: fixed-order reduction of per-block partials -> BN scale/shift
// ---------------------------------------------------------------------------
__global__ void stats_kernel(const float* __restrict__ partials,
                             const float* __restrict__ g, const float* __restrict__ bt,
                             float* __restrict__ scale, float* __restrict__ shift,
                             int O, int nblocks) {
    int o = threadIdx.x;
    if (o >= O) return;
    float s1 = 0.f, s2 = 0.f;
    for (int b = 0; b < nblocks; ++b) {
        s1 += partials[(size_t)b * 2 * O + o];
        s2 += partials[(size_t)b * 2 * O + O + o];
    }
    const float cnt = (float)(TOTAL_PTS * KNN);
    float mean = s1 / cnt;
    float var  = s2 / cnt - mean * mean;
    float sc = g[o] * rsqrtf(var + 1e-5f);
    scale[o] = sc;
    shift[o] = bt[o] - mean * sc;
}

// ---------------------------------------------------------------------------
// K4: h5[cloud][s][256] -> out (4, 256, 32, 256)
// ---------------------------------------------------------------------------
__global__ void out_kernel(const float* __restrict__ h5, float* __restrict__ out) {
    int tid = blockIdx.x * blockDim.x + threadIdx.x;
    if (tid >= 4 * 256 * 32 * 256) return;
    int s = tid & 255;
    int p = (tid >> 8) & 31;
    int d = (tid >> 13) & 255;
    int b = tid >> 21;
    out[tid] = h5[(((size_t)(b * 32 + p) * 256 + s) << 8) + d];
}

// ---------------------------------------------------------------------------
template <int C_IN, int O>
static void run_stage(const float* hin, float* hout, int* idx, float* partials,
                      float* scale, float* shift, const float* W, const float* b,
                      const float* g, const float* bt, hipStream_t stream) {
    knn_kernel<C_IN><<<NCLOUD, 256, 0, stream>>>(hin, idx);
    edge_gemm_kernel<C_IN, O, true><<<GEMM_BLOCKS, 256, 0, stream>>>(
        hin, idx, W, b, nullptr, nullptr, partials, nullptr);
    stats_kernel<<<1, 256, 0, stream>>>(partials, g, bt, scale, shift, O, GEMM_BLOCKS);
    edge_gemm_kernel<C_IN, O, false><<<GEMM_BLOCKS, 256, 0, stream>>>(
        hin, idx, W, b, scale, shift, nullptr, hout);
}

extern "C" void kernel_launch(void* const* d_in, const int* in_sizes, int n_in,
                              void* d_out, int out_size, void* d_ws, size_t ws_size,
                              hipStream_t stream) {
    const float* x = (const float*)d_in[0];
    const float *Ws[5], *bs_[5], *gs[5], *bts[5];
    for (int i = 0; i < 5; ++i) {
        Ws[i]  = (const float*)d_in[1 + 4 * i];
        bs_[i] = (const float*)d_in[2 + 4 * i];
        gs[i]  = (const float*)d_in[3 + 4 * i];
        bts[i] = (const float*)d_in[4 + 4 * i];
    }
    // workspace layout (~78 MB total)
    char* ws = (char*)d_ws;
    const size_t HMAX = (size_t)TOTAL_PTS * 256 * sizeof(float);          // 33.5 MB
    float* hA       = (float*)ws;
    float* hB       = (float*)(ws + HMAX);
    int*   idx      = (int*)  (ws + 2 * HMAX);                            // 2 MB
    float* partials = (float*)(ws + 2 * HMAX + (size_t)TOTAL_PTS * KNN * 4); // 8 MB
    float* scale    = (float*)((char*)partials + (size_t)GEMM_BLOCKS * 512 * 4);
    float* shift    = scale + 256;

    prep_kernel<<<(4 * 32 * 256 * 6 + 255) / 256, 256, 0, stream>>>(x, hA);

    run_stage<6, 16>   (hA, hB, idx, partials, scale, shift, Ws[0], bs_[0], gs[0], bts[0], stream);
    run_stage<16, 32>  (hB, hA, idx, partials, scale, shift, Ws[1], bs_[1], gs[1], bts[1], stream);
    run_stage<32, 64>  (hA, hB, idx, partials, scale, shift, Ws[2], bs_[2], gs[2], bts[2], stream);
    run_stage<64, 128> (hB, hA, idx, partials, scale, shift, Ws[3], bs_[3], gs[3], bts[3], stream);
    run_stage<128, 256>(hA, hB, idx, partials, scale, shift, Ws[4], bs_[4], gs[4], bts[4], stream);

    out_kernel<<<(4 * 256 * 32 * 256 + 255) / 256, 256, 0, stream>>>(hB, (float*)d_out);
}